// Model_37847251813189
// MI455X (gfx1250) — compile-verified
//
#include <hip/hip_runtime.h>
#include <math.h>

// ---------------------------------------------------------------------------
// GAT (2-layer, PyG-style) forward for MI455X / gfx1250.
//   layer1: xh1 = x @ W1  (WMMA f32 16x16x4, A tile staged in LDS),
//           attn softmax over incoming edges (3 edge passes:
//           max / sum / scatter), relu(out/den + b1)
//   layer2: xh2 = h1 @ W2 (WMMA), attn, head-mean + b2, global max pool,
//           classifier GEMV.
// Precision: all f32 (V_WMMA_F32_16X16X4_F32) — the net is L2/atomic bound,
// so full precision is free. Segment-max uses an order-preserving u32 key
// (monotone f32->u32 map) + atomicMax_u32; aggregation uses atomicAdd_f32.
// GEMM: one block = one 16-row slab of C; 8 waves each own a 16x16 tile
// along N. The shared 16xK A tile is staged once into LDS (coalesced),
// rows padded to K+4 floats so WMMA A-fragment reads are bank-conflict-free
// ds_load_b64s (lanes 0-15 -> banks {4l,4l+1}, lanes 16-31 -> {4l+2,4l+3}).
// ---------------------------------------------------------------------------

typedef __attribute__((ext_vector_type(2))) float v2f;
typedef __attribute__((ext_vector_type(8))) float v8f;

#define NEG_SLOPE 0.2f

__device__ __forceinline__ unsigned f2key(float f) {
  unsigned u = __float_as_uint(f);
  return (u & 0x80000000u) ? ~u : (u | 0x80000000u);
}
__device__ __forceinline__ float key2f(unsigned k) {
  return __uint_as_float((k & 0x80000000u) ? (k & 0x7FFFFFFFu) : ~k);
}

// ---------------- fill ----------------
__global__ void fill_u32(unsigned* __restrict__ p, unsigned v, long n) {
  long i = (long)blockIdx.x * blockDim.x + threadIdx.x;
  if (i < n) p[i] = v;
}

// ---------------- GEMM: C[M,Nn] = A[M,K] @ B[K,Nn] via f32 WMMA --------------
// Grid: M/16 blocks x 256 threads (8 waves). Nn == 128 (8 tiles of 16).
// M % 16 == 0, K % 4 == 0. Dynamic LDS: 16*(K+4) floats.
__global__ void gemm_wmma_f32(const float* __restrict__ A,
                              const float* __restrict__ B,
                              float* __restrict__ C, int M, int K, int Nn) {
  extern __shared__ float Alds[];       // [16][K+4]
  const int stride = K + 4;
  const int m0 = blockIdx.x << 4;

  // ---- cooperative, coalesced stage of the 16xK A tile into LDS ----
  for (int i = threadIdx.x; i < (K << 4); i += blockDim.x) {
    int r = i / K, c = i % K;
    Alds[r * stride + c] = A[(size_t)(m0 + r) * K + c];
  }
  __syncthreads();

  int wave = threadIdx.x >> 5;
  int lane = threadIdx.x & 31;
  int n0 = wave << 4;
  if (n0 >= Nn) return;                 // wave-uniform: EXEC stays all-ones

  int half = lane >> 4;                 // 0: K={k0,k0+1}  1: K={k0+2,k0+3}
  int l = lane & 15;                    // A: row index, B: col index

  v8f acc = {};
  const float* Arow = Alds + l * stride;
  const float* Bcol = B + n0 + l;
  for (int k0 = 0; k0 < K; k0 += 4) {
    int ka = k0 + 2 * half;             // ka even -> 8B-aligned LDS read
    v2f a = *(const v2f*)(Arow + ka);   // ds_load_b64, conflict-free
    v2f b;
    b.x = Bcol[(size_t)ka * Nn];
    b.y = Bcol[(size_t)(ka + 1) * Nn];
    acc = __builtin_amdgcn_wmma_f32_16x16x4_f32(
        /*neg_a=*/false, a, /*neg_b=*/false, b,
        /*c_mod=*/(short)0, acc, /*reuse_a=*/false, /*reuse_b=*/false);
  }
  // C/D layout: VGPR r -> row m0 + r + 8*half, col n0 + l
  float* Cout = C + (size_t)(m0 + 8 * half) * Nn + n0 + l;
#pragma unroll
  for (int r = 0; r < 8; ++r) Cout[(size_t)r * Nn] = acc[r];
}

// ---------------- attention coefficients: als/ald[n,h] = <xh[n,h,:], a[h,:]> ----
__global__ void attn_coeff(const float* __restrict__ xh,
                           const float* __restrict__ a_src,
                           const float* __restrict__ a_dst,
                           float* __restrict__ als, float* __restrict__ ald,
                           int Nn, int H, int Cc) {
  long t = (long)blockIdx.x * blockDim.x + threadIdx.x;
  if (t >= (long)Nn * H) return;
  int h = (int)(t % H);
  long n = t / H;
  const float* xr = xh + n * (long)(H * Cc) + (long)h * Cc;
  const float* as = a_src + h * Cc;
  const float* ad = a_dst + h * Cc;
  float s = 0.f, d = 0.f;
  for (int c = 0; c < Cc; ++c) {
    float v = xr[c];
    s += v * as[c];
    d += v * ad[c];
  }
  als[t] = s;
  ald[t] = d;
}

// ---------------- edge pass 1: segment max of leaky(als[s]+ald[d]) over dst ----
__global__ void edge_max(const int* __restrict__ ei, int E, int Nn, int H,
                         const float* __restrict__ als,
                         const float* __restrict__ ald,
                         unsigned* __restrict__ mkey) {
  long e = (long)blockIdx.x * blockDim.x + threadIdx.x;
  long ET = (long)E + Nn;
  if (e >= ET) return;
  int s, d;
  if (e < E) { s = ei[e]; d = ei[E + e]; } else { s = d = (int)(e - E); }
  for (int h = 0; h < H; ++h) {
    float v = als[(long)s * H + h] + ald[(long)d * H + h];
    v = v > 0.f ? v : NEG_SLOPE * v;
    atomicMax(&mkey[(long)d * H + h], f2key(v));
  }
}

// ---------------- edge pass 2: denom[d,h] += exp(e - max) ----------------
__global__ void edge_sum(const int* __restrict__ ei, int E, int Nn, int H,
                         const float* __restrict__ als,
                         const float* __restrict__ ald,
                         const unsigned* __restrict__ mkey,
                         float* __restrict__ denom) {
  long e = (long)blockIdx.x * blockDim.x + threadIdx.x;
  long ET = (long)E + Nn;
  if (e >= ET) return;
  int s, d;
  if (e < E) { s = ei[e]; d = ei[E + e]; } else { s = d = (int)(e - E); }
  for (int h = 0; h < H; ++h) {
    float v = als[(long)s * H + h] + ald[(long)d * H + h];
    v = v > 0.f ? v : NEG_SLOPE * v;
    float m = key2f(mkey[(long)d * H + h]);
    atomicAdd(&denom[(long)d * H + h], expf(v - m));
  }
}

// ---------------- edge pass 3: num[d,h,c] += exp(e-max) * xh[s,h,c] -----------
// 32 lanes per edge; HC == H*Cc channels strided by 32 per lane.
__global__ void edge_scatter(const int* __restrict__ ei, int E, int Nn, int H,
                             int Cc, const float* __restrict__ als,
                             const float* __restrict__ ald,
                             const unsigned* __restrict__ mkey,
                             const float* __restrict__ xh,
                             float* __restrict__ num) {
  long t = (long)blockIdx.x * blockDim.x + threadIdx.x;
  int lane = (int)(t & 31);
  long e = t >> 5;
  long ET = (long)E + Nn;
  if (e >= ET) return;
  int s, d;
  if (e < E) { s = ei[e]; d = ei[E + e]; } else { s = d = (int)(e - E); }
  int HC = H * Cc;
  const float* xs = xh + (long)s * HC;
  float* nd = num + (long)d * HC;
  for (int j = lane; j < HC; j += 32) {
    int h = j / Cc;
    float v = als[(long)s * H + h] + ald[(long)d * H + h];
    v = v > 0.f ? v : NEG_SLOPE * v;
    float m = key2f(mkey[(long)d * H + h]);
    float w = expf(v - m);
    atomicAdd(&nd[j], w * xs[j]);
  }
}

// ---------------- layer1 finalize (in place): h = relu(num/den + b1) ----------
__global__ void fin1(float* __restrict__ num, const float* __restrict__ denom,
                     const float* __restrict__ b, int Nn, int H, int Cc) {
  int HC = H * Cc;
  long t = (long)blockIdx.x * blockDim.x + threadIdx.x;
  if (t >= (long)Nn * HC) return;
  int j = (int)(t % HC);
  long n = t / HC;
  int h = j / Cc;
  float v = num[t] / (denom[n * H + h] + 1e-16f) + b[j];
  num[t] = v > 0.f ? v : 0.f;
}

// ---------------- layer2 finalize + global max pool (key-encoded) -------------
__global__ void fin2_pool(const float* __restrict__ num,
                          const float* __restrict__ denom,
                          const float* __restrict__ b2,
                          const int* __restrict__ batch,
                          unsigned* __restrict__ poolkey, int Nn, int OUTC,
                          int H) {
  long t = (long)blockIdx.x * blockDim.x + threadIdx.x;
  if (t >= (long)Nn * OUTC) return;
  int c = (int)(t % OUTC);
  long n = t / OUTC;
  float acc = 0.f;
  for (int h = 0; h < H; ++h)
    acc += num[n * (long)(H * OUTC) + (long)h * OUTC + c] /
           (denom[n * H + h] + 1e-16f);
  float v = acc / (float)H + b2[c];
  atomicMax(&poolkey[(long)batch[n] * OUTC + c], f2key(v));
}

// ---------------- classifier: out[g,k] = bc[k] + sum_c pooled[g,c]*Wc[c,k] ----
__global__ void classify(const unsigned* __restrict__ poolkey,
                         const float* __restrict__ Wc,
                         const float* __restrict__ bc, float* __restrict__ out,
                         int Gg, int OUTC, int NCc) {
  int t = blockIdx.x * blockDim.x + threadIdx.x;
  if (t >= Gg * NCc) return;
  int k = t % NCc;
  int g = t / NCc;
  float acc = bc[k];
  for (int c = 0; c < OUTC; ++c)
    acc += key2f(poolkey[g * OUTC + c]) * Wc[c * NCc + k];
  out[t] = acc;
}

// ---------------------------------------------------------------------------
extern "C" void kernel_launch(void* const* d_in, const int* in_sizes, int n_in,
                              void* d_out, int out_size, void* d_ws,
                              size_t ws_size, hipStream_t stream) {
  (void)n_in; (void)ws_size;
  const float* x      = (const float*)d_in[0];
  const int*   ei     = (const int*)d_in[1];
  const int*   batch  = (const int*)d_in[2];
  const float* W1     = (const float*)d_in[3];
  const float* a_src1 = (const float*)d_in[4];
  const float* a_dst1 = (const float*)d_in[5];
  const float* b1     = (const float*)d_in[6];
  const float* W2     = (const float*)d_in[7];
  const float* a_src2 = (const float*)d_in[8];
  const float* a_dst2 = (const float*)d_in[9];
  const float* b2     = (const float*)d_in[10];
  const float* Wc     = (const float*)d_in[11];
  const float* bc     = (const float*)d_in[12];
  float* out = (float*)d_out;

  const int N = in_sizes[2];       // 100000
  const int E = in_sizes[1] / 2;   // 1600000
  const int IN = in_sizes[0] / N;  // 64
  const int H1 = 4, C1 = 32, H2 = 2, C2 = 64;
  const int HC = 128;              // H1*C1 == H2*C2 == 128
  const int NC = 26;
  const int G = out_size / NC;     // 128
  const long ET = (long)E + N;     // edges + self loops

  // ---- workspace carve-up (256B aligned) ----
  char* w = (char*)d_ws;
  size_t off = 0;
  auto carve = [&](size_t bytes) {
    void* p = w + off;
    off += (bytes + 255) & ~(size_t)255;
    return p;
  };
  float*    xh      = (float*)carve((size_t)N * HC * 4);  // xh1, then xh2
  float*    num     = (float*)carve((size_t)N * HC * 4);  // accum -> h1 -> accum2
  float*    als     = (float*)carve((size_t)N * H1 * 4);
  float*    ald     = (float*)carve((size_t)N * H1 * 4);
  unsigned* mkey    = (unsigned*)carve((size_t)N * H1 * 4);
  float*    denom   = (float*)carve((size_t)N * H1 * 4);
  unsigned* poolkey = (unsigned*)carve((size_t)G * C2 * 4);

  const int BT = 256;
  auto blocks = [&](long n) { return (unsigned)((n + BT - 1) / BT); };

  // ================= layer 1 =================
  fill_u32<<<blocks((long)N * HC), BT, 0, stream>>>((unsigned*)num, 0u, (long)N * HC);
  fill_u32<<<blocks((long)N * H1), BT, 0, stream>>>(mkey, 0u, (long)N * H1);
  fill_u32<<<blocks((long)N * H1), BT, 0, stream>>>((unsigned*)denom, 0u, (long)N * H1);

  gemm_wmma_f32<<<N / 16, 256, (size_t)16 * (IN + 4) * 4, stream>>>(
      x, W1, xh, N, IN, HC);
  attn_coeff<<<blocks((long)N * H1), BT, 0, stream>>>(xh, a_src1, a_dst1, als, ald, N, H1, C1);
  edge_max<<<blocks(ET), BT, 0, stream>>>(ei, E, N, H1, als, ald, mkey);
  edge_sum<<<blocks(ET), BT, 0, stream>>>(ei, E, N, H1, als, ald, mkey, denom);
  edge_scatter<<<blocks(ET * 32), BT, 0, stream>>>(ei, E, N, H1, C1, als, ald, mkey, xh, num);
  fin1<<<blocks((long)N * HC), BT, 0, stream>>>(num, denom, b1, N, H1, C1);

  // ================= layer 2 =================
  gemm_wmma_f32<<<N / 16, 256, (size_t)16 * (HC + 4) * 4, stream>>>(
      num /*h1*/, W2, xh, N, HC, HC);
  // h1 (in num) is dead after gemm2 -> re-zero accumulators
  fill_u32<<<blocks((long)N * HC), BT, 0, stream>>>((unsigned*)num, 0u, (long)N * HC);
  fill_u32<<<blocks((long)N * H2), BT, 0, stream>>>(mkey, 0u, (long)N * H2);
  fill_u32<<<blocks((long)N * H2), BT, 0, stream>>>((unsigned*)denom, 0u, (long)N * H2);
  fill_u32<<<blocks((long)G * C2), BT, 0, stream>>>(poolkey, 0u, (long)G * C2);

  attn_coeff<<<blocks((long)N * H2), BT, 0, stream>>>(xh, a_src2, a_dst2, als, ald, N, H2, C2);
  edge_max<<<blocks(ET), BT, 0, stream>>>(ei, E, N, H2, als, ald, mkey);
  edge_sum<<<blocks(ET), BT, 0, stream>>>(ei, E, N, H2, als, ald, mkey, denom);
  edge_scatter<<<blocks(ET * 32), BT, 0, stream>>>(ei, E, N, H2, C2, als, ald, mkey, xh, num);
  fin2_pool<<<blocks((long)N * C2), BT, 0, stream>>>(num, denom, b2, batch, poolkey, N, C2, H2);

  // ================= classifier =================
  classify<<<blocks((long)G * NC), BT, 0, stream>>>(poolkey, Wc, bc, out, G, C2, NC);
}